// EquivariantGNN_6141803233970
// MI455X (gfx1250) — compile-verified
//
#include <hip/hip_runtime.h>

typedef __attribute__((ext_vector_type(2))) float v2f;
typedef __attribute__((ext_vector_type(8))) float v8f;

#define NN   50000
#define EE   600000
#define GG   512
#define HID  128

// ---------------------------------------------------------------------------
// Build padded input features: x16[n][0:3]=pos, [3:13]=one_hot(z), [13:16]=0
// ---------------------------------------------------------------------------
__global__ void build_x16_kernel(const float* __restrict__ pos,
                                 const int* __restrict__ z,
                                 float* __restrict__ x16) {
    int idx = blockIdx.x * blockDim.x + threadIdx.x;   // over N*16
    if (idx >= NN * 16) return;
    int n = idx >> 4;
    int c = idx & 15;
    float v = 0.0f;
    if (c < 3)       v = pos[n * 3 + c];
    else if (c < 13) v = (z[n] == (c - 3)) ? 1.0f : 0.0f;
    x16[idx] = v;
}

// ---------------------------------------------------------------------------
// WMMA f32 GEMM:  C[N x 128] = act(A[N x K] @ W[KW x 128] + bias)
// One block (256 thr = 8 waves) per 16-row tile; wave w owns 16-col tile w.
// A tile staged in LDS (row pad +4 to avoid bank conflicts).
// K, KW compile-time -> fully unrolled chain of v_wmma_f32_16x16x4_f32.
// ---------------------------------------------------------------------------
template<int K, int KW, bool RELU>
__global__ void __launch_bounds__(256)
gemm_wmma_kernel(const float* __restrict__ A,
                 const float* __restrict__ W,
                 const float* __restrict__ bias,
                 float* __restrict__ C) {
    constexpr int LSTR = K + 4;
    __shared__ float As[16 * LSTR];

    const int row0 = blockIdx.x * 16;
    const int tid  = threadIdx.x;

    // cooperative load of 16 x K A-tile (contiguous in global)
    const float* Ab = A + (size_t)row0 * K;
    #pragma unroll
    for (int i = tid; i < 16 * K; i += 256) {
        int r  = i / K;
        int cc = i - r * K;
        As[r * LSTR + cc] = Ab[i];
    }
    __syncthreads();

    const int wave = tid >> 5;          // 0..7  -> column tile
    const int lane = tid & 31;
    const int col0 = wave * 16;
    const int m    = lane & 15;         // row within tile (A) / col within tile (B,D)
    const int kh   = (lane >> 4) * 2;   // lanes 16-31 hold K+2,K+3

    v8f acc = {};
    #pragma unroll
    for (int k = 0; k < K; k += 4) {
        v2f a;
        a.x = As[m * LSTR + k + kh];
        a.y = As[m * LSTR + k + kh + 1];
        v2f b;
        int kb = k + kh;
        b.x = (kb     < KW) ? W[(size_t)(kb    ) * HID + col0 + m] : 0.0f;
        b.y = (kb + 1 < KW) ? W[(size_t)(kb + 1) * HID + col0 + m] : 0.0f;
        acc = __builtin_amdgcn_wmma_f32_16x16x4_f32(
                  false, a, false, b, (short)0, acc, false, false);
    }

    const int   col = col0 + m;
    const float bv  = bias[col];
    const int   rsel = (lane >> 4) * 8;     // D layout: vgpr i -> rows i / i+8
    #pragma unroll
    for (int i = 0; i < 8; ++i) {
        float v = acc[i] + bv;
        if (RELU) v = v > 0.0f ? v : 0.0f;
        C[(size_t)(row0 + i + rsel) * HID + col] = v;
    }
}

// ---------------------------------------------------------------------------
// agg = relu(xl)   (self-loop message, also initializes scatter target)
// ---------------------------------------------------------------------------
__global__ void init_agg_kernel(const float* __restrict__ xl,
                                float* __restrict__ agg) {
    int idx = blockIdx.x * blockDim.x + threadIdx.x;   // over N*128
    if (idx >= NN * HID) return;
    float v = xl[idx];
    agg[idx] = v > 0.0f ? v : 0.0f;
}

// ---------------------------------------------------------------------------
// Fused edge pass: ea = edge_attr@w_edge + b_edge (w_edge in LDS),
//                  msg = relu(xl[src] + ea), agg[dst] += msg (fp32 atomics).
// 128 threads = one channel each; EPB edges per block.
// ---------------------------------------------------------------------------
#define EPB 8
__global__ void __launch_bounds__(128)
edge_msg_kernel(const int* __restrict__ ei,          // [2, E]
                const float* __restrict__ eattr,     // [E, 16]
                const float* __restrict__ w_edge,    // [16, 128]
                const float* __restrict__ b_edge,    // [128]
                const float* __restrict__ xl,        // [N, 128]
                float* __restrict__ agg) {           // [N, 128]
    __shared__ float wl[16 * HID];
    __shared__ float bl[HID];
    const int c = threadIdx.x;
    #pragma unroll
    for (int i = c; i < 16 * HID; i += 128) wl[i] = w_edge[i];
    bl[c] = b_edge[c];
    __syncthreads();

    const int e0 = blockIdx.x * EPB;
    #pragma unroll 1
    for (int j = 0; j < EPB; ++j) {
        const int e   = e0 + j;
        const int src = ei[e];
        const int dst = ei[EE + e];
        const float* er = eattr + (size_t)e * 16;
        float acc = bl[c];
        #pragma unroll
        for (int k = 0; k < 16; ++k) acc += er[k] * wl[k * HID + c];
        float mv = xl[(size_t)src * HID + c] + acc;
        mv = mv > 0.0f ? mv : 0.0f;
        atomicAdd(agg + (size_t)dst * HID + c, mv);
    }
}

// ---------------------------------------------------------------------------
// Pooling: pooled[batch[n]] += x[n]   (pooled zeroed first)
// ---------------------------------------------------------------------------
__global__ void zero_kernel(float* __restrict__ p, int n) {
    int idx = blockIdx.x * blockDim.x + threadIdx.x;
    if (idx < n) p[idx] = 0.0f;
}

__global__ void pool_kernel(const float* __restrict__ x,
                            const int* __restrict__ batch,
                            float* __restrict__ pooled) {
    int idx = blockIdx.x * blockDim.x + threadIdx.x;   // over N*128
    if (idx >= NN * HID) return;
    int n = idx >> 7;
    int c = idx & 127;
    atomicAdd(pooled + (size_t)batch[n] * HID + c, x[idx]);
}

// ---------------------------------------------------------------------------
// Readout: out[g] = pooled[g] . w_out + b_out
// ---------------------------------------------------------------------------
__global__ void readout_kernel(const float* __restrict__ pooled,
                               const float* __restrict__ w_out,
                               const float* __restrict__ b_out,
                               float* __restrict__ out) {
    int g = blockIdx.x * blockDim.x + threadIdx.x;
    if (g >= GG) return;
    float s = b_out[0];
    #pragma unroll 4
    for (int c = 0; c < HID; ++c) s += pooled[(size_t)g * HID + c] * w_out[c];
    out[g] = s;
}

// ---------------------------------------------------------------------------
extern "C" void kernel_launch(void* const* d_in, const int* in_sizes, int n_in,
                              void* d_out, int out_size, void* d_ws, size_t ws_size,
                              hipStream_t stream) {
    const float* pos        = (const float*)d_in[0];
    const int*   z          = (const int*)  d_in[1];
    const int*   edge_index = (const int*)  d_in[2];
    const float* edge_attr  = (const float*)d_in[3];
    const int*   batch      = (const int*)  d_in[4];
    const float* w_node1 = (const float*)d_in[5];  const float* b_node1 = (const float*)d_in[6];
    const float* w_edge1 = (const float*)d_in[7];  const float* b_edge1 = (const float*)d_in[8];
    const float* w_mlp1a = (const float*)d_in[9];  const float* b_mlp1a = (const float*)d_in[10];
    const float* w_mlp1b = (const float*)d_in[11]; const float* b_mlp1b = (const float*)d_in[12];
    const float* w_node2 = (const float*)d_in[13]; const float* b_node2 = (const float*)d_in[14];
    const float* w_edge2 = (const float*)d_in[15]; const float* b_edge2 = (const float*)d_in[16];
    const float* w_mlp2a = (const float*)d_in[17]; const float* b_mlp2a = (const float*)d_in[18];
    const float* w_mlp2b = (const float*)d_in[19]; const float* b_mlp2b = (const float*)d_in[20];
    const float* w_out   = (const float*)d_in[21]; const float* b_out   = (const float*)d_in[22];
    float* out = (float*)d_out;

    // workspace layout (floats)
    float* ws     = (float*)d_ws;
    float* x16    = ws;                        // N*16
    float* xl     = x16  + (size_t)NN * 16;    // N*128
    float* agg    = xl   + (size_t)NN * HID;   // N*128
    float* h      = agg  + (size_t)NN * HID;   // N*128
    float* xbuf   = h    + (size_t)NN * HID;   // N*128
    float* pooled = xbuf + (size_t)NN * HID;   // G*128

    const int rowTiles = NN / 16;                       // 3125
    const int nElem    = NN * HID;                      // 6.4M
    const int eBlocks  = EE / EPB;                      // 75000

    // ---- input features ----
    build_x16_kernel<<<(NN * 16 + 255) / 256, 256, 0, stream>>>(pos, z, x16);

    // ================= layer 1 =================
    gemm_wmma_kernel<16, 13, false><<<rowTiles, 256, 0, stream>>>(x16, w_node1, b_node1, xl);
    init_agg_kernel<<<(nElem + 255) / 256, 256, 0, stream>>>(xl, agg);
    edge_msg_kernel<<<eBlocks, 128, 0, stream>>>(edge_index, edge_attr, w_edge1, b_edge1, xl, agg);
    gemm_wmma_kernel<128, 128, true ><<<rowTiles, 256, 0, stream>>>(agg, w_mlp1a, b_mlp1a, h);
    gemm_wmma_kernel<128, 128, true ><<<rowTiles, 256, 0, stream>>>(h,   w_mlp1b, b_mlp1b, xbuf); // incl. outer relu

    // ================= layer 2 =================
    gemm_wmma_kernel<128, 128, false><<<rowTiles, 256, 0, stream>>>(xbuf, w_node2, b_node2, xl);
    init_agg_kernel<<<(nElem + 255) / 256, 256, 0, stream>>>(xl, agg);
    edge_msg_kernel<<<eBlocks, 128, 0, stream>>>(edge_index, edge_attr, w_edge2, b_edge2, xl, agg);
    gemm_wmma_kernel<128, 128, true ><<<rowTiles, 256, 0, stream>>>(agg, w_mlp2a, b_mlp2a, h);
    gemm_wmma_kernel<128, 128, true ><<<rowTiles, 256, 0, stream>>>(h,   w_mlp2b, b_mlp2b, xbuf);

    // ---- pooling + readout ----
    zero_kernel<<<(GG * HID + 255) / 256, 256, 0, stream>>>(pooled, GG * HID);
    pool_kernel<<<(nElem + 255) / 256, 256, 0, stream>>>(xbuf, batch, pooled);
    readout_kernel<<<(GG + 255) / 256, 256, 0, stream>>>(pooled, w_out, b_out, out);
}